// LSTMClassifier_61572651155541
// MI455X (gfx1250) — compile-verified
//
#include <hip/hip_runtime.h>
#include <math.h>
#include <stdint.h>

// ---------------------------------------------------------------------------
// LSTM classifier on MI455X (gfx1250).
//
// Roofline: ~412 GFLOP fp32 total; weights (25 MB) are L2-resident (192 MB),
// HBM (23.3 TB/s) is irrelevant -> the limit is the 512-step serial
// recurrence, i.e. per-step *latency*. Design consequences:
//  * One fused GEMM per step, K = E+H = 1536:
//      gates^T[4096,64] = [W_ih | W_hh] x [emb(x_t) ; h]^T
//  * K split into 4 deterministic partial sums -> 1024 concurrent waves/step,
//    per-wave critical path = 384 WMMAs (vs 1536 unsplit).
//  * B operand staged global->LDS with GLOBAL_LOAD_ASYNC_TO_LDS_B128
//    (ASYNCcnt), double-buffered so the copy hides behind the WMMAs.
//  * fp32 WMMA (V_WMMA_F32_16X16X4_F32): reference is fp32 and errors
//    compound over 512 recurrent steps; bf16 would drift.
//
// WMMA mapping (ISA 7.12.2): lane l, VGPR j -> K = k0 + 2*(l>>4) + j for both
// A and B fragments (one 8B float2 per fragment); C/D: VGPR r holds
// (M = r + 8*(l>>4), N = l&15).
// ---------------------------------------------------------------------------

#define B_     64
#define T_     512
#define E_     512
#define H_     1024
#define G4_    4096          // 4*H gate rows (PyTorch order i,f,g,o)
#define KTOT_  1536          // E + H
#define NCLS_  5
#define NSPLIT 4             // deterministic K-split (partial-sum buffers)
#define KS_    (KTOT_ / NSPLIT)   // 384, multiple of KC; 512 boundary is chunk-aligned
#define LDS_STRIDE 36        // dwords; rows 16B-aligned, bank-conflict-free frag reads
#define KC     32            // K chunk staged in LDS per iteration

typedef __attribute__((ext_vector_type(2))) float v2f;
typedef __attribute__((ext_vector_type(8))) float v8f;

// 64B contiguous global->LDS async copy (4x b128), no VGPR round-trip.
// INST_OFFSET applies to both the LDS and global address (ISA 08 §4.4).
__device__ __forceinline__ void async_copy64B(unsigned lds_off, const float* gsrc) {
  asm volatile(
      "global_load_async_to_lds_b128 %0, %1, off\n\t"
      "global_load_async_to_lds_b128 %0, %1, off offset:16\n\t"
      "global_load_async_to_lds_b128 %0, %1, off offset:32\n\t"
      "global_load_async_to_lds_b128 %0, %1, off offset:48"
      :: "v"(lds_off), "v"((unsigned long long)(uintptr_t)gsrc)
      : "memory");
}

__device__ __forceinline__ void wait_async0() {
#if __has_builtin(__builtin_amdgcn_s_wait_asynccnt)
  __builtin_amdgcn_s_wait_asynccnt(0);
#else
  asm volatile("s_wait_asynccnt 0x0" ::: "memory");
#endif
}

// Per-timestep fused gates GEMM, one K-split partial per block-group:
//   gates[part][g][b] = sum_{k in part} W[g][k] * xcat[b][k]
// Block: 128 threads = 4 waves; wave = one 16-gate tile x all 64 batches.
__global__ __launch_bounds__(128) void lstm_gates_wmma(
    const int* __restrict__ x, const float* __restrict__ emb,
    const float* __restrict__ W_ih, const float* __restrict__ W_hh,
    const float* __restrict__ h_prev, float* __restrict__ gates, int t)
{
  __shared__ float ldsX[2][B_ * LDS_STRIDE];   // double-buffered xcat chunk

  const int tid  = threadIdx.x;
  const int lane = tid & 31;
  const int wave = tid >> 5;
  const int lm   = lane & 15;        // N / M index within tile
  const int half = lane >> 4;        // K sub-offset selector

  const int part = blockIdx.x >> 6;  // K-split id (0..3)
  const int gblk = blockIdx.x & 63;  // gate-tile block (4 tiles/block)
  const int kbeg = part * KS_;

  // Staging role: thread -> (batch row, 64B segment). emb row 0 is already
  // all-zero in the input (padding_idx), matching the reference.
  const int brow = tid >> 1;
  const int seg  = tid & 1;
  const int tok  = x[brow * T_ + t];
  const float* erow = emb    + (size_t)tok  * E_;
  const float* hrow = h_prev + (size_t)brow * H_;

  const int g0 = (gblk * 4 + wave) * 16;
  const float* wih = W_ih + (size_t)(g0 + lm) * E_;
  const float* whh = W_hh + (size_t)(g0 + lm) * H_;

  // LDS byte offsets for this thread's 64B segment in each buffer
  // (generic shared pointer low 32 bits == LDS offset on gfx1250).
  const unsigned lds0 = (unsigned)(uintptr_t)&ldsX[0][brow * LDS_STRIDE + seg * 16];
  const unsigned lds1 = (unsigned)(uintptr_t)&ldsX[1][brow * LDS_STRIDE + seg * 16];

  // Prologue: async-stage chunk 0 into buffer 0.
  {
    const float* src = (kbeg < E_) ? (erow + kbeg) : (hrow + (kbeg - E_));
    async_copy64B(lds0, src + seg * 16);
  }
  wait_async0();
  __syncthreads();

  v8f acc0 = {}, acc1 = {}, acc2 = {}, acc3 = {};
  const int ko = 2 * half;
  int p = 0;
  for (int k0 = kbeg; k0 < kbeg + KS_; k0 += KC) {
    // Prefetch next chunk into the other buffer (overlaps the WMMAs below).
    const int kn = k0 + KC;
    if (kn < kbeg + KS_) {
      const float* src = (kn < E_) ? (erow + kn) : (hrow + (kn - E_));
      async_copy64B(p ? lds0 : lds1, src + seg * 16);
    }

    const float* arow = (k0 < E_) ? (wih + k0) : (whh + (k0 - E_));
    const float* xb = &ldsX[p][0];
#pragma unroll
    for (int kk = 0; kk < KC; kk += 4) {
      v2f a  = *(const v2f*)(arow + kk + ko);                       // weights: L2-hot
      v2f b0 = *(const v2f*)(xb + (lm     ) * LDS_STRIDE + kk + ko); // LDS, conflict-free
      v2f b1 = *(const v2f*)(xb + (lm + 16) * LDS_STRIDE + kk + ko);
      v2f b2 = *(const v2f*)(xb + (lm + 32) * LDS_STRIDE + kk + ko);
      v2f b3 = *(const v2f*)(xb + (lm + 48) * LDS_STRIDE + kk + ko);
      // 4 independent fp32 WMMAs: pipelined, A fragment reused 4x.
      acc0 = __builtin_amdgcn_wmma_f32_16x16x4_f32(false, a, false, b0, (short)0, acc0, false, false);
      acc1 = __builtin_amdgcn_wmma_f32_16x16x4_f32(false, a, false, b1, (short)0, acc1, false, false);
      acc2 = __builtin_amdgcn_wmma_f32_16x16x4_f32(false, a, false, b2, (short)0, acc2, false, false);
      acc3 = __builtin_amdgcn_wmma_f32_16x16x4_f32(false, a, false, b3, (short)0, acc3, false, false);
    }

    wait_async0();      // my prefetch landed
    __syncthreads();    // everyone's prefetch landed / reads done
    p ^= 1;
  }

  // Store this K-split's partial D tiles (lanes 0..15 write 64B rows).
  float* gpart = gates + (size_t)part * (G4_ * B_);
  const int mrow = g0 + 8 * half;
#pragma unroll
  for (int r = 0; r < 8; ++r) {
    float* orow = gpart + (size_t)(mrow + r) * B_ + lm;
    orow[0]  = acc0[r];
    orow[16] = acc1[r];
    orow[32] = acc2[r];
    orow[48] = acc3[r];
  }
}

// Pointwise LSTM cell: fixed-order partial-sum reduction + gates + max-pool.
__global__ __launch_bounds__(256) void lstm_cell(
    const float* __restrict__ gates, const float* __restrict__ b_ih,
    const float* __restrict__ b_hh, float* __restrict__ h,
    float* __restrict__ c, float* __restrict__ feat)
{
  const int idx = blockIdx.x * blockDim.x + threadIdx.x;  // 0..B*H-1
  const int b = idx & (B_ - 1);   // coalesced over gates rows
  const int j = idx >> 6;         // hidden index (uniform per 64-thread group)
  float g4[4];
#pragma unroll
  for (int q = 0; q < 4; ++q) {
    const int row = q * H_ + j;
    float s = b_ih[row] + b_hh[row];
#pragma unroll
    for (int pp = 0; pp < NSPLIT; ++pp)
      s += gates[(size_t)pp * (G4_ * B_) + (size_t)row * B_ + b];
    g4[q] = s;
  }
  const float si = 1.0f / (1.0f + expf(-g4[0]));
  const float sf = 1.0f / (1.0f + expf(-g4[1]));
  const float tg = tanhf(g4[2]);
  const float so = 1.0f / (1.0f + expf(-g4[3]));
  const size_t o = (size_t)b * H_ + j;
  const float cn = sf * c[o] + si * tg;
  const float hn = so * tanhf(cn);
  c[o] = cn;
  h[o] = hn;
  feat[o] = fmaxf(feat[o], hn);
}

__global__ void lstm_init(float* h, float* c, float* feat) {
  const int i = blockIdx.x * blockDim.x + threadIdx.x;
  if (i < B_ * H_) { h[i] = 0.0f; c[i] = 0.0f; feat[i] = -3.402823466e38f; }
}

// Tiny FC: one wave per (class, batch), wave32 shuffle reduction.
__global__ __launch_bounds__(32) void lstm_fc(
    const float* __restrict__ feat, const float* __restrict__ W_fc,
    const float* __restrict__ b_fc, float* __restrict__ out)
{
  const int cls = blockIdx.x, b = blockIdx.y, lane = threadIdx.x;
  const float* fr = feat + (size_t)b * H_;
  const float* wr = W_fc + (size_t)cls * H_;
  float s = 0.0f;
  for (int j = lane; j < H_; j += 32) s += fr[j] * wr[j];
#pragma unroll
  for (int off = 16; off > 0; off >>= 1) s += __shfl_xor(s, off, 32);
  if (lane == 0) out[b * NCLS_ + cls] = s + b_fc[cls];
}

extern "C" void kernel_launch(void* const* d_in, const int* in_sizes, int n_in,
                              void* d_out, int out_size, void* d_ws, size_t ws_size,
                              hipStream_t stream) {
  const int*   x    = (const int*)  d_in[0];
  const float* emb  = (const float*)d_in[1];
  const float* W_ih = (const float*)d_in[2];
  const float* W_hh = (const float*)d_in[3];
  const float* b_ih = (const float*)d_in[4];
  const float* b_hh = (const float*)d_in[5];
  const float* W_fc = (const float*)d_in[6];
  const float* b_fc = (const float*)d_in[7];
  float* out = (float*)d_out;

  // Workspace (fp32): h[64,1024] c[64,1024] feat[64,1024]
  // gates[NSPLIT][4096][64]  -> 1,245,184 floats = 4.75 MB.
  float* h     = (float*)d_ws;
  float* c     = h    + B_ * H_;
  float* feat  = c    + B_ * H_;
  float* gates = feat + B_ * H_;

  lstm_init<<<(B_ * H_ + 255) / 256, 256, 0, stream>>>(h, c, feat);

  for (int t = 0; t < T_; ++t) {
    // 256 blocks x 4 waves = 1024 waves/step (256 gate tiles x 4 K-splits)
    lstm_gates_wmma<<<64 * NSPLIT, 128, 0, stream>>>(x, emb, W_ih, W_hh, h, gates, t);
    lstm_cell<<<(B_ * H_) / 256, 256, 0, stream>>>(gates, b_ih, b_hh, h, c, feat);
  }

  lstm_fc<<<dim3(NCLS_, B_), 32, 0, stream>>>(feat, W_fc, b_fc, out);
}